// BasicAttention_28570122453300
// MI455X (gfx1250) — compile-verified
//
#include <hip/hip_runtime.h>
#include <math.h>

typedef __attribute__((ext_vector_type(2))) float v2f;
typedef __attribute__((ext_vector_type(8))) float v8f;

#define B_  8
#define S1_ 64
#define S2_ 512
#define DH_ 1024
#define DF_ 1024
#define DM_ 512

#if __has_builtin(__builtin_amdgcn_tanhf)
#define TANHF __builtin_amdgcn_tanhf
#elif __has_builtin(__builtin_amdgcn_tanh_f32)
#define TANHF __builtin_amdgcn_tanh_f32
#else
#define TANHF tanhf
#endif

static __device__ __forceinline__ v8f wmma_f32(v2f a, v2f b, v8f c) {
  // D = A(16x4,f32) * B(4x16,f32) + C(16x16,f32)
  return __builtin_amdgcn_wmma_f32_16x16x4_f32(
      /*neg_a=*/false, a, /*neg_b=*/false, b,
      /*c_mod=*/(short)0, c, /*reuse_a=*/false, /*reuse_b=*/false);
}

// ---------------------------------------------------------------------------
// Kernel 1: h_proj[bs][m] = hidden[bs][:] . Wh[m][:] + bh[m]
//   M = B*S1 = 512 rows, N = DM = 512 cols, K = DH = 1024.
//   A lane loads {hidden[row][k..k+1]}   (contiguous pair)
//   B lane loads {Wh[col][k..k+1]}       (contiguous pair, since Wh is [m][d])
// ---------------------------------------------------------------------------
__global__ void __launch_bounds__(256)
hproj_kernel(const float* __restrict__ hidden, const float* __restrict__ Wh,
             const float* __restrict__ bh, float* __restrict__ hp) {
  const int wave = threadIdx.x >> 5;
  const int lane = threadIdx.x & 31;
  const int half = lane >> 4;   // 0: K pair k..k+1 | rows r..r+7 ; 1: k+2..k+3 | rows r+8..
  const int l    = lane & 15;

  const int tile = blockIdx.x * 8 + wave;        // 0..1023  (32x32 tiles)
  const int i0 = (tile >> 5) * 16;               // output row block (bs)
  const int j0 = (tile & 31) * 16;               // output col block (m)

  const float* arow = hidden + (size_t)(i0 + l) * DH_ + 2 * half;
  const float* brow = Wh     + (size_t)(j0 + l) * DH_ + 2 * half;

  v8f c = {};
#pragma unroll 8
  for (int k = 0; k < DH_; k += 4) {
    v2f a = *(const v2f*)(arow + k);
    v2f b = *(const v2f*)(brow + k);
    c = wmma_f32(a, b, c);
  }
  const float bias = bh[j0 + l];
#pragma unroll
  for (int r = 0; r < 8; ++r)
    hp[(size_t)(i0 + r + 8 * half) * DM_ + j0 + l] = c[r] + bias;
}

// ---------------------------------------------------------------------------
// Kernel 2: fpT[b][m][t] = Wf[m][:] . feats[b][t][:] + bf[m]   (transposed proj)
//   Per batch: M = DM = 512 (m), N = S2 = 512 (t), K = DF = 1024.
//   Output stored t-contiguous so the scores kernel streams coalesced lines.
// ---------------------------------------------------------------------------
__global__ void __launch_bounds__(256)
fprojT_kernel(const float* __restrict__ feats, const float* __restrict__ Wf,
              const float* __restrict__ bf, float* __restrict__ fpT) {
  const int wave = threadIdx.x >> 5;
  const int lane = threadIdx.x & 31;
  const int half = lane >> 4;
  const int l    = lane & 15;

  const int idx = blockIdx.x * 8 + wave;         // 0..8191
  const int b   = idx >> 10;                     // /1024 tiles per batch
  const int t   = idx & 1023;
  const int i0 = (t >> 5) * 16;                  // m block
  const int j0 = (t & 31) * 16;                  // t block

  const float* arow = Wf + (size_t)(i0 + l) * DF_ + 2 * half;
  const float* brow = feats + (size_t)b * S2_ * DF_ + (size_t)(j0 + l) * DF_ + 2 * half;

  v8f c = {};
#pragma unroll 8
  for (int k = 0; k < DF_; k += 4) {
    v2f a = *(const v2f*)(arow + k);
    v2f bm = *(const v2f*)(brow + k);
    c = wmma_f32(a, bm, c);
  }
  float* out = fpT + (size_t)b * DM_ * S2_;
#pragma unroll
  for (int r = 0; r < 8; ++r) {
    const int m = i0 + r + 8 * half;
    out[(size_t)m * S2_ + j0 + l] = c[r] + bf[m];
  }
}

// ---------------------------------------------------------------------------
// Kernel 3: per (b,s): scores[t] = sum_m v[m]*tanh(hp[bs][m] + fpT[b][m][t]),
// then softmax over t (S2=512). One block (256 thr) per (b,s); thread owns
// t = tid and t = tid+256. fpT reads are lane-contiguous; hp/v are LDS
// broadcasts. tanh goes to the TRANS pipe (v_tanh_f32).
// ---------------------------------------------------------------------------
__global__ void __launch_bounds__(256)
scores_softmax_kernel(const float* __restrict__ hp, const float* __restrict__ fpT,
                      const float* __restrict__ v, float* __restrict__ weight) {
  __shared__ float hp_s[DM_];
  __shared__ float v_s[DM_];
  __shared__ float red[256];

  const int tid = threadIdx.x;
  const int bs  = blockIdx.x;                    // 0..511
  const int b   = bs >> 6;                       // / S1

  const float* hprow = hp + (size_t)bs * DM_;
  hp_s[tid]       = hprow[tid];
  hp_s[tid + 256] = hprow[tid + 256];
  v_s[tid]        = v[tid];
  v_s[tid + 256]  = v[tid + 256];
  __syncthreads();

  const float* fb = fpT + (size_t)b * DM_ * S2_;
  float acc0 = 0.0f, acc1 = 0.0f;
#pragma unroll 4
  for (int m = 0; m < DM_; ++m) {
    const float f0 = fb[(size_t)m * S2_ + tid];
    const float f1 = fb[(size_t)m * S2_ + tid + 256];
    const float h  = hp_s[m];
    const float vv = v_s[m];
    acc0 = fmaf(vv, TANHF(h + f0), acc0);
    acc1 = fmaf(vv, TANHF(h + f1), acc1);
  }

  // block-wide max
  red[tid] = fmaxf(acc0, acc1);
  __syncthreads();
#pragma unroll
  for (int off = 128; off > 0; off >>= 1) {
    if (tid < off) red[tid] = fmaxf(red[tid], red[tid + off]);
    __syncthreads();
  }
  const float mx = red[0];
  __syncthreads();

  const float e0 = __expf(acc0 - mx);
  const float e1 = __expf(acc1 - mx);
  red[tid] = e0 + e1;
  __syncthreads();
#pragma unroll
  for (int off = 128; off > 0; off >>= 1) {
    if (tid < off) red[tid] += red[tid + off];
    __syncthreads();
  }
  const float inv = 1.0f / red[0];

  float* wrow = weight + (size_t)bs * S2_;
  wrow[tid]       = e0 * inv;
  wrow[tid + 256] = e1 * inv;
}

// ---------------------------------------------------------------------------
// Kernel 4: context[b] = weight[b] (64x512) @ feats[b] (512x1024), WMMA f32.
//   A pair loads contiguous (weight k-contig); B pair strided by DF (rows k, k+1).
// ---------------------------------------------------------------------------
__global__ void __launch_bounds__(256)
context_kernel(const float* __restrict__ weight, const float* __restrict__ feats,
               float* __restrict__ context) {
  const int wave = threadIdx.x >> 5;
  const int lane = threadIdx.x & 31;
  const int half = lane >> 4;
  const int l    = lane & 15;

  const int idx = blockIdx.x * 8 + wave;         // 0..2047
  const int b   = idx >> 8;                      // 256 tiles per batch (4 x 64)
  const int t   = idx & 255;
  const int i0 = (t >> 6) * 16;                  // s block
  const int j0 = (t & 63) * 16;                  // d block

  const float* arow = weight + (size_t)b * S1_ * S2_ + (size_t)(i0 + l) * S2_ + 2 * half;
  const float* fb   = feats + (size_t)b * S2_ * DF_;

  v8f c = {};
#pragma unroll 4
  for (int k = 0; k < S2_; k += 4) {
    v2f a = *(const v2f*)(arow + k);
    const float* bp = fb + (size_t)(k + 2 * half) * DF_ + j0 + l;
    v2f bm;
    bm.x = bp[0];
    bm.y = bp[DF_];
    c = wmma_f32(a, bm, c);
  }
  float* out = context + (size_t)b * S1_ * DF_;
#pragma unroll
  for (int r = 0; r < 8; ++r)
    out[(size_t)(i0 + r + 8 * half) * DF_ + j0 + l] = c[r];
}

// ---------------------------------------------------------------------------
extern "C" void kernel_launch(void* const* d_in, const int* in_sizes, int n_in,
                              void* d_out, int out_size, void* d_ws, size_t ws_size,
                              hipStream_t stream) {
  const float* hidden = (const float*)d_in[0];   // [B,S1,DH]
  const float* feats  = (const float*)d_in[1];   // [B,S2,DF]
  const float* Wh     = (const float*)d_in[2];   // [DM,DH]
  const float* bh     = (const float*)d_in[3];   // [DM]
  const float* Wf     = (const float*)d_in[4];   // [DM,DF]
  const float* bf     = (const float*)d_in[5];   // [DM]
  const float* v      = (const float*)d_in[6];   // [DM]

  float* context = (float*)d_out;                         // B*S1*DF = 524288
  float* weight  = context + (size_t)B_ * S1_ * DF_;      // B*S1*S2 = 262144

  float* hp  = (float*)d_ws;                              // B*S1*DM  (1 MB)
  float* fpT = hp + (size_t)B_ * S1_ * DM_;               // B*DM*S2  (8 MB)

  hproj_kernel<<<128, 256, 0, stream>>>(hidden, Wh, bh, hp);
  fprojT_kernel<<<1024, 256, 0, stream>>>(feats, Wf, bf, fpT);
  scores_softmax_kernel<<<512, 256, 0, stream>>>(hp, fpT, v, weight);
  context_kernel<<<256, 256, 0, stream>>>(weight, feats, context);
}